// MultiHeadAttention_5884105195636
// MI455X (gfx1250) — compile-verified
//
#include <hip/hip_runtime.h>
#include <hip/hip_bf16.h>
#include <math.h>

// Problem dims (compile-time)
#define BB   4
#define SS   512
#define FF   512
#define HH   8
#define DD   64
#define RR   64
#define HD   512   // H*D
#define OO   512   // OUT

#define KC   64          // K-chunk staged per double-buffer step
#define BST  66          // padded LDS stride per column (KC + 2, kills bank conflicts)
#define BUFN (16 * BST)  // floats per buffer

typedef __attribute__((ext_vector_type(2))) float v2f;
typedef __attribute__((ext_vector_type(8))) float v8f;

// One V_WMMA_F32_16X16X4_F32 step. A fragment: lane holds A[m][kbase+koff .. +1],
// m = lane&15, koff = 2*(lane>>4).  B fragment: lane holds B[kbase+koff..+1][n], n = lane&15.
__device__ __forceinline__ v8f wmma4(v2f a, v2f b, v8f acc) {
  return __builtin_amdgcn_wmma_f32_16x16x4_f32(false, a, false, b, (short)0, acc,
                                               false, false);
}

// CDNA5 async global->LDS copy (ASYNCcnt-tracked, no VGPR data).
// %0 = 32-bit LDS byte address, %1 = 64-bit global address.
__device__ __forceinline__ void async_g2l_b32(unsigned lds_byte, const float* g) {
  asm volatile("global_load_async_to_lds_b32 %0, %1, off"
               :: "v"(lds_byte), "v"(g) : "memory");
}
__device__ __forceinline__ void wait_async_le8() {
  asm volatile("s_wait_asynccnt 0x8" ::: "memory");
}
__device__ __forceinline__ void wait_async_0() {
  asm volatile("s_wait_asynccnt 0x0" ::: "memory");
}

// ---------------------------------------------------------------------------
// Batched f32 WMMA GEMM, LDS-staged B panel:  C = alpha * (A @ B) + bias
//   A element (m,k) = A[m*lda + k]            (row major)
//   B element (k,n) = B[k*ldbr + n*ldbc]      (arbitrary strides -> handles B^T)
//   batch z: base += (z/bmod)*s1 + (z%bmod)*s2   for A, B, C independently
// Block = 128 (4 waves). Workgroup tile = 64 M x 16 N; wave w owns rows
// [64*blockIdx.y + 16w, +16). B panel (KC x 16) is async-copied to LDS
// transposed+padded, double buffered; each wave reads fragments via ds_load_b64.
// grid: x = N/16, y = M/64, z = batches.  Requires K % KC == 0, M % 64 == 0.
// ---------------------------------------------------------------------------
__global__ void gemm_wmma_f32(const float* __restrict__ A, int lda, int sA1, int sA2,
                              const float* __restrict__ Bm, int ldbr, int ldbc,
                              int sB1, int sB2,
                              float* __restrict__ C, int ldc, int sC1, int sC2,
                              const float* __restrict__ bias, int K, int bmod,
                              float alpha) {
  __shared__ float bbuf[2][BUFN];

  const int z = blockIdx.z;
  const float* Ab = A  + (z / bmod) * sA1 + (z % bmod) * sA2;
  const float* Bb = Bm + (z / bmod) * sB1 + (z % bmod) * sB2;
  float*       Cb = C  + (z / bmod) * sC1 + (z % bmod) * sC2;

  const int n0   = blockIdx.x * 16;
  const int t    = threadIdx.x;        // 0..127
  const int wv   = t >> 5;             // wave id 0..3
  const int lane = t & 31;
  const int mn   = lane & 15;          // A-frag row / B-frag col within tile
  const int koff = (lane >> 4) << 1;   // 0 or 2
  const int m0   = blockIdx.y * 64 + wv * 16;

  // cooperative async-copy assignment: thread t covers (n = t&15, k = t>>4 + 8j)
  const int tn = t & 15;
  const int tk = t >> 4;               // 0..7
  const unsigned ldsbase = (unsigned)(size_t)&bbuf[0][0];

  // per-lane pointers
  const float* Ap = Ab + (size_t)(m0 + mn) * lda;
  for (int k = 0; k < K; k += 64) __builtin_prefetch(Ap + k, 0, 1);  // global_prefetch

  // prologue: stage chunk 0 into buffer 0 (8 async b32 issues per thread)
  {
    const float* gp = Bb + (size_t)tk * ldbr + (size_t)(n0 + tn) * ldbc;
    unsigned lp = ldsbase + (unsigned)(tn * BST + tk) * 4u;
#pragma unroll
    for (int j = 0; j < 8; ++j) {
      async_g2l_b32(lp, gp);
      gp += (size_t)8 * ldbr;
      lp += 8u * 4u;
    }
  }

  v8f acc = {};
  for (int kc = 0; kc < K; kc += KC) {
    const int cur = (kc / KC) & 1;
    if (kc + KC < K) {   // stage next chunk into the other buffer, then drain current
      const float* gp = Bb + (size_t)(kc + KC + tk) * ldbr + (size_t)(n0 + tn) * ldbc;
      unsigned lp = ldsbase + (unsigned)((cur ^ 1) * BUFN + tn * BST + tk) * 4u;
#pragma unroll
      for (int j = 0; j < 8; ++j) {
        async_g2l_b32(lp, gp);
        gp += (size_t)8 * ldbr;
        lp += 8u * 4u;
      }
      wait_async_le8();            // current chunk's 8 issues retired (in-order)
    } else {
      wait_async_0();
    }
    __syncthreads();               // all 4 waves' panel stores visible

    const float* Ac = Ap + kc;
    const float* Bc = &bbuf[cur][mn * BST];
#pragma unroll
    for (int kk = 0; kk < KC; kk += 4) {
      v2f a = *(const v2f*)(Ac + kk + koff);          // global_load_b64
      v2f b = *(const v2f*)(Bc + kk + koff);          // ds_load_b64 (padded, no conflict)
      acc = wmma4(a, b, acc);
    }
    __syncthreads();               // protect buffer reuse two chunks later
  }

  const int n     = n0 + mn;
  const float bc  = bias ? bias[n] : 0.0f;
  const int rbase = m0 + ((lane >> 4) << 3);   // C/D: vgpr i -> row i (+8 upper half-wave)
#pragma unroll
  for (int i = 0; i < 8; ++i)
    Cb[(size_t)(rbase + i) * ldc + n] = acc[i] * alpha + bc;
}

// ---------------------------------------------------------------------------
// scores[b,h,q,k] = (q . k  +  p[b,h,q, adj[b,q,k]]) * scale , masked -> -inf
// written directly into the attn region of d_out.
// grid: x = S/16 (key tile), y = S/16 (query tile), z = B*H; block = 32
// ---------------------------------------------------------------------------
__global__ void scores_wmma(const float* __restrict__ qws,   // [B][S][HD]
                            const float* __restrict__ kws,   // [B][S][HD]
                            const float* __restrict__ pws,   // [B][H][S][R]
                            const int*   __restrict__ adj,   // [B][S][S]
                            const unsigned char* __restrict__ mask, // [B][S]
                            float* __restrict__ attn,        // [B][H][S][S]
                            float scale) {
  const int z = blockIdx.z, b = z / HH, h = z % HH;
  const float* Qb = qws + b * SS * HD + h * DD;
  const float* Kb = kws + b * SS * HD + h * DD;

  const int m0   = blockIdx.y * 16;
  const int n0   = blockIdx.x * 16;
  const int lane = threadIdx.x & 31;
  const int mn   = lane & 15;
  const int koff = (lane >> 4) << 1;

  const float* Ap = Qb + (m0 + mn) * HD;   // query row
  const float* Bp = Kb + (n0 + mn) * HD;   // B = K^T : (d,n) -> K[n][d], contiguous in d

  v8f acc = {};
#pragma unroll
  for (int k = 0; k < DD; k += 4) {
    v2f a = *(const v2f*)(Ap + k + koff);
    v2f bf = *(const v2f*)(Bp + k + koff);
    acc = wmma4(a, bf, acc);
  }

  const int n      = n0 + mn;                        // key index
  const bool mk    = mask[b * SS + n] != 0;
  const int rbase  = m0 + ((lane >> 4) << 3);
  const float* Pb  = pws + z * SS * RR;
  const int* adjb  = adj + b * SS * SS;
  float* out       = attn + z * SS * SS;
#pragma unroll
  for (int i = 0; i < 8; ++i) {
    const int q = rbase + i;
    const int r = adjb[q * SS + n];                  // relative-position bucket
    const float v = (acc[i] + Pb[q * RR + r]) * scale;
    out[q * SS + n] = mk ? -INFINITY : v;
  }
}

// ---------------------------------------------------------------------------
// Row softmax (in place in attn) + bucket accumulation w[row][r] = sum over k
// of attn where adj==r.  grid = B*H*S blocks of 128 threads (4 keys/thread).
// ---------------------------------------------------------------------------
__global__ void softmax_bucket(float* __restrict__ attn,       // [B*H*S][S]
                               const int* __restrict__ adj,    // [B][S][S]
                               float* __restrict__ w) {        // [B*H*S][R]
  const int row = blockIdx.x;            // (b*H+h)*S + q
  const int z   = row / SS;
  const int q   = row % SS;
  const int b   = z / HH;
  float* arow       = attn + (size_t)row * SS;
  const int* adjr   = adj + ((size_t)b * SS + q) * SS;

  __shared__ float red[128];
  __shared__ float buckets[RR];
  const int t = threadIdx.x;

  float vals[4];
  float mx = -INFINITY;
#pragma unroll
  for (int i = 0; i < 4; ++i) { vals[i] = arow[t + 128 * i]; mx = fmaxf(mx, vals[i]); }
  red[t] = mx; __syncthreads();
  for (int s = 64; s > 0; s >>= 1) {
    if (t < s) red[t] = fmaxf(red[t], red[t + s]);
    __syncthreads();
  }
  mx = red[0]; __syncthreads();

  float sum = 0.0f;
#pragma unroll
  for (int i = 0; i < 4; ++i) { vals[i] = __expf(vals[i] - mx); sum += vals[i]; }
  red[t] = sum; __syncthreads();
  for (int s = 64; s > 0; s >>= 1) {
    if (t < s) red[t] += red[t + s];
    __syncthreads();
  }
  const float inv = 1.0f / red[0];

  if (t < RR) buckets[t] = 0.0f;
  __syncthreads();
#pragma unroll
  for (int i = 0; i < 4; ++i) {
    const float a = vals[i] * inv;
    arow[t + 128 * i] = a;                      // final attention weight
    atomicAdd(&buckets[adjr[t + 128 * i]], a);  // ds_add_f32 bucket scatter
  }
  __syncthreads();
  if (t < RR) w[(size_t)row * RR + t] = buckets[t];
}

// ---------------------------------------------------------------------------
// ctx_total[b][s][h*D+d] = ctx[b][h][s][d] + (w[b][h][s][:] @ emb_v)[d]
// grid: x = D/16, y = S/16, z = B*H; block = 32
// ---------------------------------------------------------------------------
__global__ void combine_wmma(const float* __restrict__ w,       // [B][H][S][R]
                             const float* __restrict__ emb_v,   // [R][D]
                             const float* __restrict__ ctx,     // [B][H][S][D]
                             float* __restrict__ ctx_total) {   // [B][S][HD]
  const int z = blockIdx.z, b = z / HH, h = z % HH;
  const int m0   = blockIdx.y * 16;
  const int n0   = blockIdx.x * 16;
  const int lane = threadIdx.x & 31;
  const int mn   = lane & 15;
  const int koff = (lane >> 4) << 1;

  const float* Ap = w + ((size_t)z * SS + m0 + mn) * RR;
  const float* Bp = emb_v + n0 + mn;        // (r,d) -> emb_v[r*D + d]

  v8f acc = {};
#pragma unroll
  for (int k = 0; k < RR; k += 4) {
    v2f a = *(const v2f*)(Ap + k + koff);
    v2f bf;
    bf.x = Bp[(k + koff) * DD];
    bf.y = Bp[(k + koff + 1) * DD];
    acc = wmma4(a, bf, acc);
  }

  const int n     = n0 + mn;
  const int rbase = m0 + ((lane >> 4) << 3);
#pragma unroll
  for (int i = 0; i < 8; ++i) {
    const int s = rbase + i;
    const float v = acc[i] + ctx[((size_t)z * SS + s) * DD + n];
    ctx_total[((size_t)b * SS + s) * HD + h * DD + n] = v;
  }
}

// ---------------------------------------------------------------------------
extern "C" void kernel_launch(void* const* d_in, const int* in_sizes, int n_in,
                              void* d_out, int out_size, void* d_ws, size_t ws_size,
                              hipStream_t stream) {
  // inputs per setup_inputs() order
  const float* key   = (const float*)d_in[0];
  const float* value = (const float*)d_in[1];
  const float* query = (const float*)d_in[2];
  const unsigned char* mask = (const unsigned char*)d_in[3];
  const int*   adj   = (const int*)d_in[4];
  const float* Wq = (const float*)d_in[5];
  const float* bq = (const float*)d_in[6];
  const float* Wk = (const float*)d_in[7];
  const float* bk = (const float*)d_in[8];
  const float* Wv = (const float*)d_in[9];
  const float* bv = (const float*)d_in[10];
  const float* Wo = (const float*)d_in[11];
  const float* bo = (const float*)d_in[12];
  const float* emb_k = (const float*)d_in[13];
  const float* emb_v = (const float*)d_in[14];

  // outputs: out (B,S,OUT) then attn (B,H,S,S)
  float* out  = (float*)d_out;
  float* attn = (float*)d_out + (size_t)BB * SS * OO;

  // workspace layout (floats); w aliases q, ctx_total aliases k (both dead)
  const size_t M1 = (size_t)BB * SS * HD;   // 1,048,576 floats each
  float* qws = (float*)d_ws;
  float* kws = qws + M1;
  float* vws = kws + M1;
  float* pws = vws + M1;                    // [B][H][S][R]
  float* cws = pws + M1;                    // ctx [B][H][S][D]
  float* wws = qws;                         // alias: q dead after scores
  float* tws = kws;                         // alias: k dead after scores

  const float scale = 1.0f / sqrtf((float)DD);
  const dim3 wave(32, 1, 1);
  const dim3 quad(128, 1, 1);

  // 1-3: Q/K/V projections  (2048 x 512 x 512 GEMMs), 64x16 macro tiles
  dim3 gProj(HD / 16, (BB * SS) / 64, 1);
  gemm_wmma_f32<<<gProj, quad, 0, stream>>>(query, FF, 0, 0, Wq, HD, 1, 0, 0,
                                            qws, HD, 0, 0, bq, FF, 1, 1.0f);
  gemm_wmma_f32<<<gProj, quad, 0, stream>>>(key,   FF, 0, 0, Wk, HD, 1, 0, 0,
                                            kws, HD, 0, 0, bk, FF, 1, 1.0f);
  gemm_wmma_f32<<<gProj, quad, 0, stream>>>(value, FF, 0, 0, Wv, HD, 1, 0, 0,
                                            vws, HD, 0, 0, bv, FF, 1, 1.0f);

  // 4: p[b,h,s,r] = q . emb_k^T   (B element (d,r) = emb_k[r*D + d])
  dim3 gP(RR / 16, SS / 64, BB * HH);
  gemm_wmma_f32<<<gP, quad, 0, stream>>>(qws, HD, SS * HD, DD,
                                         emb_k, 1, DD, 0, 0,
                                         pws, RR, HH * SS * RR, SS * RR,
                                         nullptr, DD, HH, 1.0f);

  // 5: masked scores -> attn region of d_out
  dim3 gS(SS / 16, SS / 16, BB * HH);
  scores_wmma<<<gS, wave, 0, stream>>>(qws, kws, pws, adj, mask, attn, scale);

  // 6: softmax rows + bucket sums w
  softmax_bucket<<<dim3(BB * HH * SS, 1, 1), dim3(128, 1, 1), 0, stream>>>(attn, adj, wws);

  // 7: ctx = attn @ v   (B element (k,d) = v[b][k][h*D+d])
  dim3 gC(DD / 16, SS / 64, BB * HH);
  gemm_wmma_f32<<<gC, quad, 0, stream>>>(attn, SS, HH * SS * SS, SS * SS,
                                         vws, HD, 1, SS * HD, DD,
                                         cws, DD, HH * SS * DD, SS * DD,
                                         nullptr, SS, HH, 1.0f);

  // 8: ctx_total = ctx + w @ emb_v, transposed to [B][S][HD]
  dim3 gX(DD / 16, SS / 16, BB * HH);
  combine_wmma<<<gX, wave, 0, stream>>>(wws, emb_v, cws, tws);

  // 9: out = ctx_total @ Wo + bo
  dim3 gO(OO / 16, (BB * SS) / 64, 1);
  gemm_wmma_f32<<<gO, quad, 0, stream>>>(tws, HD, 0, 0, Wo, OO, 1, 0, 0,
                                         out, OO, 0, 0, bo, HD, 1, 1.0f);
}